// VectorQuantizer_42030549958884
// MI455X (gfx1250) — compile-verified
//
#include <hip/hip_runtime.h>
#include <hip/hip_bf16.h>

typedef __attribute__((ext_vector_type(2))) float v2f;
typedef __attribute__((ext_vector_type(8))) float v8f;

#define VQ_N      65536      // 16*64*64 flattened vectors
#define VQ_D      64
#define VQ_K      1024
#define VQ_HW     4096       // 64*64
#define VQ_ELEMS  4194304    // 16*64*64*64

// ---------------- prep: 0.5*||e_k||^2 per code, zero histogram ---------------
__global__ __launch_bounds__(1024)
void vq_prep(const float* __restrict__ E, float* __restrict__ nsq,
             unsigned int* __restrict__ counts) {
  int t = threadIdx.x;            // 1024 threads == VQ_K
  counts[t] = 0u;
  const float* row = E + t * VQ_D;
  float s = 0.f;
  #pragma unroll
  for (int d = 0; d < VQ_D; ++d) { float v = row[d]; s += v * v; }
  nsq[t] = 0.5f * s;
}

// ---------------- main: WMMA scores + argmin + quantized + loss --------------
// Grid: 512 blocks x 128 threads (4 waves). Each wave owns 32 rows.
__global__ __launch_bounds__(128)
void vq_main(const float* __restrict__ x, const float* __restrict__ E,
             const float* __restrict__ nsq,
             float* __restrict__ out_q, float* __restrict__ out_idx,
             int* __restrict__ ws_idx, float* __restrict__ wave_loss,
             unsigned int* __restrict__ counts) {
  const int lane = threadIdx.x & 31;
  const int wave = threadIdx.x >> 5;
  const int m    = lane & 15;          // row-in-tile (A) / col-in-tile (B,C)
  const int hi   = lane >> 4;         // half-wave
  const int t0   = hi ? 2 : 0;        // K sub-offset inside each 4-dim chunk
  const int gw   = blockIdx.x * 4 + wave;
  const int row0 = gw * 32;

  // ---- A fragments for two 16-row tiles: rows row0+m and row0+16+m ----
  // x is NCHW [16][64][4096]: x[n,d] = x[(n>>12)*262144 + d*4096 + (n&4095)]
  v2f a0[16], a1[16];
  {
    int n0 = row0 + m, n1 = row0 + 16 + m;
    const float* p0 = x + ((size_t)(n0 >> 12) << 18) + (n0 & 4095);
    const float* p1 = x + ((size_t)(n1 >> 12) << 18) + (n1 & 4095);
    #pragma unroll
    for (int c = 0; c < 16; ++c) {
      int d = 4 * c + t0;
      a0[c].x = p0[(size_t)d << 12];
      a0[c].y = p0[(size_t)(d + 1) << 12];
      a1[c].x = p1[(size_t)d << 12];
      a1[c].y = p1[(size_t)(d + 1) << 12];
    }
  }

  float best_v0[8], best_v1[8];
  int   best_i0[8], best_i1[8];
  #pragma unroll
  for (int r = 0; r < 8; ++r) {
    best_v0[r] = best_v1[r] = 3.4e38f;
    best_i0[r] = best_i1[r] = 0;
  }

  // ---- k-tile loop: 16 codes per tile, D=64 -> 16 chained K=4 WMMAs ----
  for (int kt = 0; kt < 64; ++kt) {
    int col = kt * 16 + m;
    float hn = nsq[col];                       // 0.5*||e_col||^2
    const float* ep = E + col * VQ_D + t0;     // 8B-aligned (t0 in {0,2})
    v2f bf[16];
    #pragma unroll
    for (int c = 0; c < 16; ++c) bf[c] = *(const v2f*)(ep + 4 * c);

    v8f acc0 = {}; v8f acc1 = {};
    #pragma unroll
    for (int c = 0; c < 16; ++c)
      acc0 = __builtin_amdgcn_wmma_f32_16x16x4_f32(false, a0[c], false, bf[c],
                                                   (short)0, acc0, false, false);
    #pragma unroll
    for (int c = 0; c < 16; ++c)
      acc1 = __builtin_amdgcn_wmma_f32_16x16x4_f32(false, a1[c], false, bf[c],
                                                   (short)0, acc1, false, false);

    // dist-equivalent score: 0.5||e||^2 - x.e  (minimize)
    #pragma unroll
    for (int r = 0; r < 8; ++r) {
      float v0 = hn - acc0[r];
      float v1 = hn - acc1[r];
      if (v0 < best_v0[r]) { best_v0[r] = v0; best_i0[r] = col; }
      if (v1 < best_v1[r]) { best_v1[r] = v1; best_i1[r] = col; }
    }
  }

  // ---- cross-lane argmin inside each 16-lane half (column classes) ----
  #pragma unroll
  for (int r = 0; r < 8; ++r) {
    #pragma unroll
    for (int off = 8; off >= 1; off >>= 1) {
      float ov = __shfl_xor(best_v0[r], off, 32);
      int   oi = __shfl_xor(best_i0[r], off, 32);
      if (ov < best_v0[r] || (ov == best_v0[r] && oi < best_i0[r])) {
        best_v0[r] = ov; best_i0[r] = oi;
      }
      ov = __shfl_xor(best_v1[r], off, 32);
      oi = __shfl_xor(best_i1[r], off, 32);
      if (ov < best_v1[r] || (ov == best_v1[r] && oi < best_i1[r])) {
        best_v1[r] = ov; best_i1[r] = oi;
      }
    }
  }

  // select best_i{0,1}[lane&7] statically (all lanes of a half agree)
  const int r8 = m & 7;
  int sel0 = 0, sel1 = 0;
  #pragma unroll
  for (int r = 0; r < 8; ++r)
    if (r8 == r) { sel0 = best_i0[r]; sel1 = best_i1[r]; }

  // ---- indices + histogram: 32 lanes cover the wave's 32 rows ----
  {
    // lo half owns tile rows 0-7, hi half rows 8-15 (per C-matrix layout)
    int myrow = (m < 8) ? (row0 + 8 * hi + r8) : (row0 + 16 + 8 * hi + r8);
    int myidx = (m < 8) ? sel0 : sel1;
    ws_idx[myrow]  = myidx;
    out_idx[myrow] = (float)myidx;
    atomicAdd(&counts[myidx], 1u);
  }

  // ---- loss partial straight from A fragments still in registers ----
  {
    int src = (m & 7) + ((m < 8) ? 0 : 16);   // half hosting tile-row m
    int i0 = __shfl(sel0, src, 32);           // code for tile0 row m
    int i1 = __shfl(sel1, src, 32);           // code for tile1 row m
    const float* e0 = E + i0 * VQ_D + t0;
    const float* e1 = E + i1 * VQ_D + t0;
    float lp = 0.f;
    #pragma unroll
    for (int c = 0; c < 16; ++c) {
      float d0 = e0[4 * c]     - a0[c].x;
      float d1 = e0[4 * c + 1] - a0[c].y;
      float d2 = e1[4 * c]     - a1[c].x;
      float d3 = e1[4 * c + 1] - a1[c].y;
      lp += d0 * d0 + d1 * d1 + d2 * d2 + d3 * d3;
    }
    #pragma unroll
    for (int off = 16; off >= 1; off >>= 1) lp += __shfl_xor(lp, off, 32);
    if (lane == 0) wave_loss[gw] = lp;        // deterministic: one slot/wave
  }

  // ---- quantized (== straight-through output) in NCHW, coalesced ----
  {
    int ml  = lane & 15;
    int src = (ml & 7) + ((ml < 8) ? 0 : 16);
    int s0  = __shfl(sel0, src, 32);
    int s1  = __shfl(sel1, src, 32);
    int idxr = (lane < 16) ? s0 : s1;         // code for row row0+lane
    int n = row0 + lane;
    float* qp = out_q + ((size_t)(n >> 12) << 18) + (n & 4095);
    const float* er = E + idxr * VQ_D;
    #pragma unroll
    for (int d = 0; d < VQ_D; ++d)
      qp[(size_t)d << 12] = er[d];            // lanes -> 128B-contiguous hw
  }
}

// ---------------- one-hot encodings: streaming nontemporal stores ------------
// One wave per row; enc base is only 8B-aligned in d_out -> f2 stores.
__global__ __launch_bounds__(256)
void vq_onehot(const int* __restrict__ ws_idx, float* __restrict__ enc) {
  int wave = threadIdx.x >> 5, lane = threadIdx.x & 31;
  int n = blockIdx.x * 8 + wave;
  int idx = ws_idx[n];
  v2f* p = (v2f*)(enc + ((size_t)n << 10));
  #pragma unroll
  for (int i = 0; i < 16; ++i) {
    int c2 = i * 32 + lane;
    int cb = c2 * 2;
    v2f v;
    v.x = (cb     == idx) ? 1.0f : 0.0f;
    v.y = (cb + 1 == idx) ? 1.0f : 0.0f;
    __builtin_nontemporal_store(v, p + c2);
  }
}

// ---------------- finalize: loss scalar + perplexity -------------------------
__global__ __launch_bounds__(1024)
void vq_final(const unsigned int* __restrict__ counts,
              const float* __restrict__ wave_loss,
              float* __restrict__ out_loss, float* __restrict__ out_perp) {
  __shared__ float sh[1024];
  __shared__ float sl[1024];
  int t = threadIdx.x;
  float p = (float)counts[t] * (1.0f / 65536.0f);
  sh[t] = -p * logf(p + 1e-10f);
  sl[t] = wave_loss[t] + wave_loss[t + 1024];
  __syncthreads();
  for (int s = 512; s > 0; s >>= 1) {
    if (t < s) { sh[t] += sh[t + s]; sl[t] += sl[t + s]; }
    __syncthreads();
  }
  if (t == 0) {
    out_perp[0] = expf(sh[0]);
    // q_latent_loss + 0.25*e_latent_loss == 1.25 * SSE / numel
    out_loss[0] = 1.25f * sl[0] * (1.0f / (float)VQ_ELEMS);
  }
}

extern "C" void kernel_launch(void* const* d_in, const int* in_sizes, int n_in,
                              void* d_out, int out_size, void* d_ws, size_t ws_size,
                              hipStream_t stream) {
  const float* x = (const float*)d_in[0];   // [16,64,64,64] NCHW f32
  const float* E = (const float*)d_in[1];   // [1024,64] f32

  float* out      = (float*)d_out;
  float* out_loss = out;                              // [1]
  float* out_q    = out + 1;                          // [4194304] NCHW
  float* out_perp = out + 1 + VQ_ELEMS;               // [1]
  float* out_enc  = out + 2 + VQ_ELEMS;               // [65536*1024]
  float* out_idx  = out + 2 + VQ_ELEMS + (size_t)VQ_N * VQ_K;  // [65536]

  char* ws = (char*)d_ws;
  int*          ws_idx = (int*)ws;                    // 65536 * 4B
  float*        nsq    = (float*)(ws + 262144);       // 1024 * 4B
  unsigned int* counts = (unsigned int*)(ws + 266240);// 1024 * 4B
  float*        wloss  = (float*)(ws + 270336);       // 2048 * 4B

  vq_prep  <<<1,    1024, 0, stream>>>(E, nsq, counts);
  vq_main  <<<512,  128,  0, stream>>>(x, E, nsq, out_q, out_idx, ws_idx,
                                       wloss, counts);
  vq_onehot<<<8192, 256,  0, stream>>>(ws_idx, out_enc);
  vq_final <<<1,    1024, 0, stream>>>(counts, wloss, out_loss, out_perp);
}